// MotionProjector_69183333204310
// MI455X (gfx1250) — compile-verified
//
#include <hip/hip_runtime.h>

typedef float v2f __attribute__((ext_vector_type(2)));
typedef float v4f __attribute__((ext_vector_type(4)));
typedef float v8f __attribute__((ext_vector_type(8)));

#define SNV (64*64*64)     // 262144 voxels
#define NB1 128            // blocks for reduce kernel
#define NW_PER_BLK 4       // waves per block (wave32)
#define NCHUNK (SNV/64)    // 4096 chunks of 64 voxels
#define NWAVES (NB1*NW_PER_BLK)        // 512
#define CHUNKS_PER_WAVE (NCHUNK/NWAVES) // 8 (compile-time constant trips)

static __device__ __forceinline__ float gcoord(int i) {
    // linspace(-1, 1, 64)[i] = (2*i - 63) / 63
    return ((float)(2 * i - 63)) * (1.0f / 63.0f);
}

// ---------------------------------------------------------------------------
// Kernel 0: zero the 128-float accumulator region of the workspace.
// ---------------------------------------------------------------------------
__global__ void k0_zero(float* __restrict__ acc) {
    if (threadIdx.x < 128) acc[threadIdx.x] = 0.0f;
}

// ---------------------------------------------------------------------------
// Kernel 1: pivot einsum  C(28x4) = Mask(28xN) x [gx,gy,gz,1](Nx4)
// computed with V_WMMA_F32_16X16X4_F32.
//   A (16x4 f32): rows = (b,k) pairs, K = 4 voxels of the current group.
//     Lane L holds A[M=L&15][K = 2*(L>>4) + {0,1}] in a v2f  (ISA 7.12.2).
//   B (4x16 f32): B[j][n] = feature n of voxel j  (n: 0=gx 1=gy 2=gz 3=1),
//     built branchlessly from per-lane selector floats so EXEC is all-ones.
//   C (16x16 f32, 8 VGPRs): row M = bk, col N = feature; accumulated across
//     all voxels this wave owns, then reduced via LDS + global float atomics.
// ---------------------------------------------------------------------------
__global__ __launch_bounds__(128) void k1_reduce(const float* __restrict__ mask,
                                                 float* __restrict__ acc_g) {
    __shared__ float tile[NW_PER_BLK][32][64];  // per-wave staged mask tile
    __shared__ float accs[128];                 // block accumulators (bk*4+f)

    const int t = threadIdx.x;
    const int w = t >> 5;   // wave in block
    const int L = t & 31;   // lane

    accs[t] = 0.0f;
    // zero pad rows 28..31 of this wave's tile (read as A rows for bk>=28)
    #pragma unroll
    for (int i = 0; i < 8; ++i)
        tile[w][28 + (L >> 3)][(L & 7) * 8 + i] = 0.0f;
    __syncthreads();

    const int mrow  = L & 15;      // A row (bk) / B col (feature) index
    const int khalf = L >> 4;      // which K-pair this lane holds
    const int klo   = khalf * 2;
    const int gwave = blockIdx.x * NW_PER_BLK + w;

    // Per-lane feature selectors (computed once; branch-free inner loop).
    const float selx = (mrow == 0) ? 1.0f : 0.0f;
    const float sely = (mrow == 1) ? 1.0f : 0.0f;
    const float selz = (mrow == 2) ? 1.0f : 0.0f;
    const float sel1 = (mrow == 3) ? 1.0f : 0.0f;
    const float gz_klo = gcoord(klo);          // z value at j = klo
    const float GZSTEP = 8.0f / 63.0f;         // delta z per group (4 voxels)
    const float GZ1    = 2.0f / 63.0f;         // delta z per voxel

    v8f c0 = {};  // bk 0..15
    v8f c1 = {};  // bk 16..27 (rows 28..31 are zero padding)

    #pragma unroll 1
    for (int it = 0; it < CHUNKS_PER_WAVE; ++it) {   // constant, uniform trips
        const int chunk = gwave + it * NWAVES;
        const int v0 = chunk * 64;  // 64-aligned -> one z-row: ix,iy constant
        // Stage 28 coalesced b64 streams into LDS: tile[w][bk][0..63]
        #pragma unroll
        for (int bk = 0; bk < 28; ++bk) {
            const int bi = bk / 7;
            const int ki = bk % 7;
            const float* p = mask + (size_t)(bi * 8 + ki) * (size_t)SNV + v0 + 2 * L;
            v2f m2 = *(const v2f*)p;
            *(v2f*)&tile[w][bk][2 * L] = m2;
        }
        const float gx = gcoord(v0 >> 12);
        const float gy = gcoord((v0 >> 6) & 63);
        const float basev = selx * gx + sely * gy + sel1;  // gz-independent part
        #pragma unroll
        for (int g = 0; g < 16; ++g) {
            const int j0 = 4 * g + klo;            // voxel-in-chunk for K=klo
            v2f a0 = *(const v2f*)&tile[w][mrow][j0];
            v2f a1 = *(const v2f*)&tile[w][16 + mrow][j0];
            const float gz0 = gz_klo + (float)g * GZSTEP;
            v2f bb;
            bb.x = __builtin_fmaf(selz, gz0, basev);
            bb.y = __builtin_fmaf(selz, gz0 + GZ1, basev);
            c0 = __builtin_amdgcn_wmma_f32_16x16x4_f32(false, a0, false, bb,
                                                       (short)0, c0, false, false);
            c1 = __builtin_amdgcn_wmma_f32_16x16x4_f32(false, a1, false, bb,
                                                       (short)0, c1, false, false);
        }
    }

    // Extract C: VGPR r holds M=r (lanes 0-15) and M=8+r (lanes 16-31), N=lane&15.
    if (mrow < 4) {
        #pragma unroll
        for (int r = 0; r < 8; ++r) {
            const int bkl = r + 8 * khalf;
            atomicAdd(&accs[bkl * 4 + mrow], c0[r]);
            atomicAdd(&accs[(16 + bkl) * 4 + mrow], c1[r]);  // bk>=28 adds 0
        }
    }
    __syncthreads();
    if (t < 112) atomicAdd(&acc_g[t], accs[t]);
}

// ---------------------------------------------------------------------------
// Kernel 2: per-(b,k) motion parameters  A = R - I,  d = p + t - R p.
// ---------------------------------------------------------------------------
__global__ void k2_params(const float* __restrict__ acc,
                          const float* __restrict__ trans_vec,
                          const float* __restrict__ rot_mat,
                          float* __restrict__ prm) {
    const int t = threadIdx.x;
    if (t >= 28) return;
    const float Sx = acc[t * 4 + 0], Sy = acc[t * 4 + 1];
    const float Sz = acc[t * 4 + 2], Sm = acc[t * 4 + 3];
    const float inv = 1.0f / Sm;
    const float px = Sx * inv, py = Sy * inv, pz = Sz * inv;
    const float* R  = rot_mat   + t * 9;   // (B,7,3,3) flat: (b*7+k)*9
    const float* tv = trans_vec + t * 3;   // (B,7,3)   flat: (b*7+k)*3
    const float r00 = R[0], r01 = R[1], r02 = R[2];
    const float r10 = R[3], r11 = R[4], r12 = R[5];
    const float r20 = R[6], r21 = R[7], r22 = R[8];
    float* P = prm + t * 12;
    P[0] = r00 - 1.0f; P[1] = r01;        P[2] = r02;
    P[3] = r10;        P[4] = r11 - 1.0f; P[5] = r12;
    P[6] = r20;        P[7] = r21;        P[8] = r22 - 1.0f;
    P[9]  = px + tv[0] - (r00 * px + r01 * py + r02 * pz);
    P[10] = py + tv[1] - (r10 * px + r11 * py + r12 * pz);
    P[11] = pz + tv[2] - (r20 * px + r21 * py + r22 * pz);
}

// ---------------------------------------------------------------------------
// Kernel 3: streaming apply.  motion[b,:,v] = sum_k mask[b,k,v]*(A_k g(v)+d_k)
// float4 per thread; k=7 contributes exactly zero and is skipped.
// ---------------------------------------------------------------------------
__global__ __launch_bounds__(256) void k3_apply(const float* __restrict__ mask,
                                                const float* __restrict__ prm,
                                                float* __restrict__ out) {
    const int b = blockIdx.y;
    const int v = (blockIdx.x * 256 + threadIdx.x) * 4;   // 4 consecutive z
    const int iz = v & 63;
    const float gx = gcoord(v >> 12);
    const float gy = gcoord((v >> 6) & 63);
    v4f gz;
    gz.x = gcoord(iz);     gz.y = gcoord(iz + 1);
    gz.z = gcoord(iz + 2); gz.w = gcoord(iz + 3);

    v4f mx = {}, my = {}, mz = {};
    const float* P = prm + b * 7 * 12;
    const float* M = mask + (size_t)(b * 8) * (size_t)SNV + v;
    #pragma unroll
    for (int k = 0; k < 7; ++k, P += 12) {
        v4f m4 = *(const v4f*)(M + (size_t)k * (size_t)SNV);
        const float sx = P[0] * gx + P[1] * gy + P[9];
        const float sy = P[3] * gx + P[4] * gy + P[10];
        const float sz = P[6] * gx + P[7] * gy + P[11];
        v4f yx = P[2] * gz + sx;
        v4f yy = P[5] * gz + sy;
        v4f yz = P[8] * gz + sz;
        mx += m4 * yx;  my += m4 * yy;  mz += m4 * yz;
    }
    float* o = out + (size_t)b * 3 * (size_t)SNV + v;
    *(v4f*)(o)            = mx;
    *(v4f*)(o + SNV)      = my;
    *(v4f*)(o + 2 * SNV)  = mz;
}

// ---------------------------------------------------------------------------
extern "C" void kernel_launch(void* const* d_in, const int* in_sizes, int n_in,
                              void* d_out, int out_size, void* d_ws, size_t ws_size,
                              hipStream_t stream) {
    (void)in_sizes; (void)n_in; (void)out_size; (void)ws_size;
    const float* mask      = (const float*)d_in[0];  // (4,8,64,64,64) f32
    const float* trans_vec = (const float*)d_in[1];  // (4,7,3) f32
    const float* rot_mat   = (const float*)d_in[2];  // (4,7,3,3) f32
    // d_in[3] = grids: recomputed analytically, not read.
    float* out = (float*)d_out;                      // (4,3,64,64,64) f32

    float* ws  = (float*)d_ws;
    float* acc = ws;          // 128 floats: [bk*4 + {Sx,Sy,Sz,Sm}]
    float* prm = ws + 128;    // 336 floats: 28 x {A(9), d(3)}

    k0_zero  <<<1,   128, 0, stream>>>(acc);
    k1_reduce<<<NB1, 128, 0, stream>>>(mask, acc);
    k2_params<<<1,    32, 0, stream>>>(acc, trans_vec, rot_mat, prm);
    k3_apply <<<dim3(SNV / 1024, 4), 256, 0, stream>>>(mask, prm, out);
}